// GINModule_13889924235831
// MI455X (gfx1250) — compile-verified
//
#include <hip/hip_runtime.h>
#include <hip/hip_bf16.h>

typedef __attribute__((ext_vector_type(2))) float v2f;
typedef __attribute__((ext_vector_type(8))) float v8f;

#define CDIM 1024
#define NDIM 1024
#define ODIM 2048
#define KNN  20

// D(M x Nc) = A^T * B where A is stored (KD x M) row-major (LDA), B is (KD x Nc)
// row-major (LDB). One wave computes a 16x64 strip: 1 A-fragment reused over 4
// B-fragments -> 4x v_wmma_f32_16x16x4_f32 per k-step.
// All leading dims / trip counts are template constants so row strides fold into
// immediate global_load offsets and the k-loop advances by constant pointer adds
// (no per-iteration v_mul_u64 address math).
// A-frag layout (ISA 7.12.2, 32-bit A 16x4): lane half h (=lane>>4) holds K=2h+{0,1}
// in VGPR {0,1}, M = lane&15. B symmetric with N = lane&15.
// D layout: VGPR r -> row 16*mTile + r + 8*half, col = lane&15.
template <int LDA, int LDB, int LDD, int KD, int MTILES>
__global__ __launch_bounds__(256) void wmma_gemm_kt(const float* __restrict__ A,
                                                    const float* __restrict__ B,
                                                    float* __restrict__ D) {
    const int lane   = threadIdx.x & 31;
    const int wave   = blockIdx.x * (blockDim.x >> 5) + (threadIdx.x >> 5);
    const int mTile  = wave % MTILES;
    const int nStrip = wave / MTILES;
    const int half = lane >> 4;
    const int l16  = lane & 15;
    const int mIdx  = mTile * 16 + l16;
    const int nBase = nStrip * 64;

    const float* __restrict__ Ap = A + (size_t)(2 * half) * LDA + mIdx;
    const float* __restrict__ Bp = B + (size_t)(2 * half) * LDB + nBase + l16;

    v8f acc0 = {}, acc1 = {}, acc2 = {}, acc3 = {};

#pragma unroll 4
    for (int kk = 0; kk < KD; kk += 4) {
        v2f a;  a.x  = Ap[0];   a.y  = Ap[LDA];
        v2f b0; b0.x = Bp[0];   b0.y = Bp[LDB];
        v2f b1; b1.x = Bp[16];  b1.y = Bp[LDB + 16];
        v2f b2; b2.x = Bp[32];  b2.y = Bp[LDB + 32];
        v2f b3; b3.x = Bp[48];  b3.y = Bp[LDB + 48];

        acc0 = __builtin_amdgcn_wmma_f32_16x16x4_f32(false, a, false, b0, (short)0, acc0, false, false);
        acc1 = __builtin_amdgcn_wmma_f32_16x16x4_f32(false, a, false, b1, (short)0, acc1, false, false);
        acc2 = __builtin_amdgcn_wmma_f32_16x16x4_f32(false, a, false, b2, (short)0, acc2, false, false);
        acc3 = __builtin_amdgcn_wmma_f32_16x16x4_f32(false, a, false, b3, (short)0, acc3, false, false);

        Ap += 4 * LDA;
        Bp += 4 * LDB;
    }

    const int rowBase = mTile * 16 + 8 * half;
    float* __restrict__ Dp = D + (size_t)rowBase * LDD + nBase + l16;
#pragma unroll
    for (int r = 0; r < 8; ++r) {
        Dp[0]  = acc0[r];
        Dp[16] = acc1[r];
        Dp[32] = acc2[r];
        Dp[48] = acc3[r];
        Dp += LDD;
    }
}

__global__ __launch_bounds__(256) void diag_kernel(const float* __restrict__ G,
                                                   float* __restrict__ diag) {
    const int i = blockIdx.x * 256 + threadIdx.x;
    diag[i] = G[(size_t)i * NDIM + i];
}

// One wave per row: dist[row][j] = 2*G[row][j] - diag[row] - diag[j].
// dist[row][row] == 0 exactly -> self is always top-1, as in the reference.
// 20 rounds of (lane-local argmax over 32 regs) + shfl_xor tree reduce,
// lower-index tie break to match jax.lax.top_k.
__global__ __launch_bounds__(256) void topk_kernel(const float* __restrict__ G,
                                                   const float* __restrict__ diag,
                                                   int* __restrict__ idxOut) {
    const int lane = threadIdx.x & 31;
    const int row  = blockIdx.x * 8 + (threadIdx.x >> 5);
    const float di = diag[row];

    float vals[32];
#pragma unroll
    for (int t = 0; t < 32; ++t) {
        const int j = t * 32 + lane;
        vals[t] = 2.0f * G[(size_t)row * NDIM + j] - di - diag[j];
    }

    for (int k = 0; k < KNN; ++k) {
        float best = -3.402823466e38f;
        int bestIdx = 0x7fffffff;
#pragma unroll
        for (int t = 0; t < 32; ++t) {
            const int j = t * 32 + lane;
            if (vals[t] > best) { best = vals[t]; bestIdx = j; }
        }
#pragma unroll
        for (int off = 16; off > 0; off >>= 1) {
            const float ov = __shfl_xor(best, off, 32);
            const int   oi = __shfl_xor(bestIdx, off, 32);
            if (ov > best || (ov == best && oi < bestIdx)) { best = ov; bestIdx = oi; }
        }
        if (lane == 0) idxOut[row * KNN + k] = bestIdx;
#pragma unroll
        for (int t = 0; t < 32; ++t) {
            if (t * 32 + lane == bestIdx) vals[t] = -3.402823466e38f;
        }
    }
}

// h[c][n] = (1 + eps[c][n]) * x[c][n] + sum_k x[c][idx[n][k]]
__global__ __launch_bounds__(256) void gin_h_kernel(const float* __restrict__ x,
                                                    const float* __restrict__ eps,
                                                    const int* __restrict__ idx,
                                                    float* __restrict__ h) {
    const int n = blockIdx.x * 256 + threadIdx.x;
    const int c = blockIdx.y;
    const float* xr = x + (size_t)c * NDIM;
    const int* ir = idx + n * KNN;
    float s = 0.0f;
#pragma unroll
    for (int k = 0; k < KNN; ++k) s += xr[ir[k]];
    h[(size_t)c * NDIM + n] = (1.0f + eps[(size_t)c * NDIM + n]) * xr[n] + s;
}

// W (O x C) -> WT (C x O), LDS-tiled so both sides stay coalesced.
__global__ __launch_bounds__(256) void transpose_kernel(const float* __restrict__ W,
                                                        float* __restrict__ WT) {
    __shared__ float tile[32][33];
    const int tx = threadIdx.x & 31;
    const int ty = threadIdx.x >> 5; // 0..7
    const int cBase = blockIdx.x * 32;
    const int oBase = blockIdx.y * 32;
#pragma unroll
    for (int r = 0; r < 32; r += 8)
        tile[ty + r][tx] = W[(size_t)(oBase + ty + r) * CDIM + cBase + tx];
    __syncthreads();
#pragma unroll
    for (int r = 0; r < 32; r += 8)
        WT[(size_t)(cBase + ty + r) * ODIM + oBase + tx] = tile[tx][ty + r];
}

extern "C" void kernel_launch(void* const* d_in, const int* in_sizes, int n_in,
                              void* d_out, int out_size, void* d_ws, size_t ws_size,
                              hipStream_t stream) {
    const float* x   = (const float*)d_in[0]; // (C, N)
    const float* w   = (const float*)d_in[1]; // (O, C, 1, 1)
    const float* eps = (const float*)d_in[2]; // (C, N, 1)
    float* out = (float*)d_out;               // (O, N)

    char* ws = (char*)d_ws;
    float* G    = (float*)(ws);                                  // 4 MB  (N*N)
    float* H    = (float*)(ws + (size_t)(4u << 20));             // 4 MB  (C*N)
    float* WT   = (float*)(ws + (size_t)(8u << 20));             // 8 MB  (C*O)
    float* diag = (float*)(ws + (size_t)(16u << 20));            // 4 KB
    int*   idx  = (int*)  (ws + (size_t)(16u << 20) + 4096);     // 80 KB (N*K)

    // 1) G = x^T x : M=1024 (64 m-tiles) x 16 n-strips = 1024 waves -> 128 blocks
    wmma_gemm_kt<NDIM, NDIM, NDIM, CDIM, 64><<<128, 256, 0, stream>>>(x, x, G);

    // 2) diag(G) == ||x_i||^2
    diag_kernel<<<NDIM / 256, 256, 0, stream>>>(G, diag);

    // 3) top-20 per row of dist = 2G - diag - diag^T
    topk_kernel<<<NDIM / 8, 256, 0, stream>>>(G, diag, idx);

    // 4) h = (1+eps)*x + neighbor-sum
    gin_h_kernel<<<dim3(NDIM / 256, CDIM), 256, 0, stream>>>(x, eps, idx, H);

    // 5) W^T staged K-major for coalesced WMMA A-fragment loads
    transpose_kernel<<<dim3(CDIM / 32, ODIM / 32), 256, 0, stream>>>(w, WT);

    // 6) out = W @ h : M=2048 (128 m-tiles) x 16 n-strips = 2048 waves -> 256 blocks
    wmma_gemm_kt<ODIM, NDIM, NDIM, CDIM, 128><<<256, 256, 0, stream>>>(WT, H, out);
}